// GraphConvolution_3401614098844
// MI455X (gfx1250) — compile-verified
//
#include <hip/hip_runtime.h>
#include <hip/hip_bf16.h>
#include <stddef.h>

// Problem constants from the reference
#define GC_B 16
#define GC_D 128
#define GC_O 64
#define GC_K 3

typedef __attribute__((ext_vector_type(2))) float v2f;
typedef __attribute__((ext_vector_type(8))) float v8f;

// ---------------------------------------------------------------------------
// Kernel 1: zero-fill the output (harness poisons d_out with 0xAA).
// out_size = B*N*K*O is divisible by 4 -> float4 fill.
// ---------------------------------------------------------------------------
__global__ void gc_zero_f32(float4* __restrict__ p, size_t n4) {
    size_t i = (size_t)blockIdx.x * blockDim.x + threadIdx.x;
    size_t stride = (size_t)gridDim.x * blockDim.x;
    const float4 z = {0.f, 0.f, 0.f, 0.f};
    for (; i < n4; i += stride) p[i] = z;
}

// ---------------------------------------------------------------------------
// Kernel 2: GEMM  pre[n,b,o] = sum_d x[b,n,d] * Wk[d,o]
// One wave per node n (16 M-rows of a WMMA tile == b=0..15 of node n).
// Wk is staged in LDS in PAIR-INTERLEAVED layout:
//   sW2[p*64 + col] = { Wk[2p][col], Wk[2p+1][col] }   (p = 0..63)
// so each B fragment is a single contiguous ds_load_b64.
// Inner loop: 1x global_load_b64 (A) + 4x ds_load_b64 (B) + 4x
// v_wmma_f32_16x16x4_f32, for 128 WMMAs per node.
// ---------------------------------------------------------------------------
__global__ void __launch_bounds__(256)
gc_gemm_wmma(const float* __restrict__ x,   // [B, N, D]
             const float* __restrict__ Wk,  // [D, O] for this support
             float* __restrict__ pre,       // [N, B, O]
             int N)
{
    __shared__ float2 sW2[(GC_D / 2) * GC_O];  // 32 KB, pair-interleaved

    // Cooperative staged repack: 64 pairs x 16 col-groups = 1024 tasks,
    // 4 per thread.  Each task: 2x b128 global load + 4x b64 LDS store.
    for (int i = threadIdx.x; i < (GC_D / 2) * (GC_O / 4); i += 256) {
        const int p = i >> 4;              // pair index: rows 2p, 2p+1
        const int g = (i & 15) << 2;       // column group start
        const float4 lo = *(const float4*)(Wk + (size_t)(2 * p) * GC_O + g);
        const float4 hi = *(const float4*)(Wk + (size_t)(2 * p + 1) * GC_O + g);
        float2* dst = sW2 + p * GC_O + g;
        dst[0] = make_float2(lo.x, hi.x);
        dst[1] = make_float2(lo.y, hi.y);
        dst[2] = make_float2(lo.z, hi.z);
        dst[3] = make_float2(lo.w, hi.w);
    }
    __syncthreads();

    const int wave = threadIdx.x >> 5;                 // 0..7
    const int lane = threadIdx.x & 31;
    const int n    = blockIdx.x * 8 + wave;            // node index
    if (n >= N) return;                                // wave-uniform: EXEC all-ones

    const int half = lane >> 4;                        // 0 or 1
    const int lid  = lane & 15;

    // A-tile row for this lane: M = lid -> b = lid.  x[b, n, :]
    const float* arow = x + (size_t)lid * (size_t)N * GC_D + (size_t)n * GC_D;

    // LDS base for this lane's B fragments (pair index advances by kk/2)
    const float2* wrow = sW2 + (size_t)half * GC_O + lid;

    v8f acc0 = {}, acc1 = {}, acc2 = {}, acc3 = {};

    #pragma unroll 4
    for (int kk = 0; kk < GC_D; kk += 4) {
        const int d0 = kk + 2 * half;      // this lane's K-values: d0, d0+1

        // A fragment (16x4 f32): v0 = A[m][2*half], v1 = A[m][2*half+1]
        v2f a = *(const v2f*)(arow + d0);  // 8B-aligned (d0 even)

        // B fragments: pair index pi = kk/2 + half; one b64 LDS load each.
        const float2* wb = wrow + (size_t)(kk >> 1) * GC_O;
        v2f b0 = *(const v2f*)(wb +  0);
        v2f b1 = *(const v2f*)(wb + 16);
        v2f b2 = *(const v2f*)(wb + 32);
        v2f b3 = *(const v2f*)(wb + 48);

        acc0 = __builtin_amdgcn_wmma_f32_16x16x4_f32(false, a, false, b0,
                                                     (short)0, acc0, false, false);
        acc1 = __builtin_amdgcn_wmma_f32_16x16x4_f32(false, a, false, b1,
                                                     (short)0, acc1, false, false);
        acc2 = __builtin_amdgcn_wmma_f32_16x16x4_f32(false, a, false, b2,
                                                     (short)0, acc2, false, false);
        acc3 = __builtin_amdgcn_wmma_f32_16x16x4_f32(false, a, false, b3,
                                                     (short)0, acc3, false, false);
    }

    // Store: D layout 16x16 f32: lane -> col = lid, VGPR j -> row M = j + 8*half.
    // pre[n, b, o] with b = M, o = ot*16 + lid   (consecutive lanes coalesce)
    float* dst = pre + (size_t)n * (GC_B * GC_O);
    #pragma unroll
    for (int j = 0; j < 8; ++j) {
        const int b = j + 8 * half;
        float* row = dst + (size_t)b * GC_O + lid;
        row[ 0] = acc0[j];
        row[16] = acc1[j];
        row[32] = acc2[j];
        row[48] = acc3[j];
    }
}

// ---------------------------------------------------------------------------
// Kernel 3: COO SpMM scatter.
// One block per edge: out[b, r, kofs+o] += v * pre[c, b, o]  (1024 floats).
// Each thread: one float4 gather (pre is L2-resident) + 4 f32 atomics.
// ---------------------------------------------------------------------------
__global__ void __launch_bounds__(256)
gc_spmm_scatter(const float* __restrict__ pre,   // [N, B, O] for this support
                const float* __restrict__ vals,  // [E]
                const int*   __restrict__ rows,  // [E]
                const int*   __restrict__ cols,  // [E]
                float* __restrict__ out,         // [B, N, K*O]
                int E, int N, int kofs)
{
    const int e = blockIdx.x;
    if (e >= E) return;
    const int   r = rows[e];
    const int   c = cols[e];
    const float v = vals[e];

    const float* src = pre + (size_t)c * (GC_B * GC_O);
    const size_t b_stride = (size_t)N * (GC_K * GC_O);   // stride between b-planes

    const int t  = threadIdx.x;        // 0..255 covers 16 b x 16 col-groups
    const int b  = t >> 4;             // 0..15
    const int og = (t & 15) << 2;      // 0,4,...,60

    const float4 s = *(const float4*)(src + (size_t)b * GC_O + og);
    float* dstp = out + (size_t)b * b_stride
                      + (size_t)r * (GC_K * GC_O) + kofs + og;
    atomicAdd(dstp + 0, v * s.x);
    atomicAdd(dstp + 1, v * s.y);
    atomicAdd(dstp + 2, v * s.z);
    atomicAdd(dstp + 3, v * s.w);
}

// ---------------------------------------------------------------------------
// Launch
// ---------------------------------------------------------------------------
extern "C" void kernel_launch(void* const* d_in, const int* in_sizes, int n_in,
                              void* d_out, int out_size, void* d_ws, size_t ws_size,
                              hipStream_t stream) {
    const float* x        = (const float*)d_in[0];   // [B, N, D]
    const float* W        = (const float*)d_in[1];   // [K, D, O]
    const float* sup_vals = (const float*)d_in[2];   // [K, E]
    const int*   sup_row  = (const int*)d_in[3];     // [K, E]
    const int*   sup_col  = (const int*)d_in[4];     // [K, E]
    float* out = (float*)d_out;                       // [B, N, K*O]

    const int N = in_sizes[0] / (GC_B * GC_D);
    const int E = in_sizes[2] / GC_K;

    float* pre = (float*)d_ws;                        // [N, B, O] ~41 MB, reused per k

    // 1) zero the output (float4 fill)
    gc_zero_f32<<<2048, 256, 0, stream>>>((float4*)out, (size_t)out_size / 4);

    // 2) per support: GEMM (WMMA) then edge scatter; sequential on stream so
    //    a single pre buffer is reused and stays L2-resident for the gather.
    const int gemm_blocks = (N + 7) / 8;              // 8 waves/block, 1 node/wave
    for (int k = 0; k < GC_K; ++k) {
        gc_gemm_wmma<<<gemm_blocks, 256, 0, stream>>>(
            x, W + (size_t)k * GC_D * GC_O, pre, N);
        gc_spmm_scatter<<<E, 256, 0, stream>>>(
            pre, sup_vals + (size_t)k * E, sup_row + (size_t)k * E,
            sup_col + (size_t)k * E, out, E, N, k * GC_O);
    }
}